// SimpleGNN_76811195121823
// MI455X (gfx1250) — compile-verified
//
#include <hip/hip_runtime.h>

// Problem constants (reference: B=4, N=96, D=512, L=2)
#define GB 4
#define GN 96
#define GD 512
#define GBN (GB * GN)          // 384 rows
#define GL 2
#define G_EPS 1e-5f
#define G_INVN (1.0f / (float)GN)

// LDS staging geometry: one workgroup = 16-row x 512-col output stripe.
#define KB 8                              // k-rows per staged weight block
#define NKB (GD / KB)                     // 64 k-blocks
#define A_LDS_FLOATS (16 * GD)            // 8192 floats = 32 KB
#define B_LDS_FLOATS (KB * GD)            // 4096 floats = 16 KB
#define LDS_TOTAL (A_LDS_FLOATS + 2 * B_LDS_FLOATS)   // 16384 floats = 64 KB

typedef float v2f __attribute__((ext_vector_type(2)));
typedef float v4f __attribute__((ext_vector_type(4)));
typedef float v8f __attribute__((ext_vector_type(8)));
typedef int   gv4i __attribute__((vector_size(16)));   // matches builtin proto

// ---------------------------------------------------------------------------
// gfx1250 async global->LDS copy (ASYNCcnt-tracked), with synchronous fallback
// Builtin proto (from clang diag): (int4 AS(1)*, int4 AS(3)*, imm i32, imm i32)
// ---------------------------------------------------------------------------
#if defined(__gfx1250__) && __has_builtin(__builtin_amdgcn_global_load_async_to_lds_b128) && __has_builtin(__builtin_amdgcn_s_wait_asynccnt)
#define ASYNC_WAIT(n) __builtin_amdgcn_s_wait_asynccnt(n)
__device__ __forceinline__ void lds_copy_f4(const float* __restrict__ g,
                                            float* __restrict__ l) {
    __builtin_amdgcn_global_load_async_to_lds_b128(
        (__attribute__((address_space(1))) gv4i*)g,
        (__attribute__((address_space(3))) gv4i*)l, 0, 0);
}
#else
#define ASYNC_WAIT(n)
__device__ __forceinline__ void lds_copy_f4(const float* __restrict__ g,
                                            float* __restrict__ l) {
    *(v4f*)l = *(const v4f*)g;
}
#endif

// ---------------------------------------------------------------------------
// Core: C[16 x 512] (+)= A[16 x 512] @ W[512 x 512] for one row stripe.
// 8 waves x (4 tiles of 16x16) per workgroup. A panel resident in LDS;
// W streamed through double-buffered LDS blocks loaded asynchronously.
// fp32 WMMA 16x16x4; A-frag: lane<16 -> K={k,k+1}, lane>=16 -> K={k+2,k+3}.
// ---------------------------------------------------------------------------
__device__ __forceinline__ void gemm_stripe(const float* __restrict__ Asrc,  // + m0*GD
                                            const float* __restrict__ W,
                                            float* __restrict__ lds,
                                            v8f acc[4]) {
    const int t    = threadIdx.x;        // 0..255
    const int wave = t >> 5;             // 0..7
    const int lane = t & 31;
    const int lr   = lane & 15;
    const int hi   = lane >> 4;
    const int n0   = wave * 64;

    float* Al = lds;
    float* B0 = lds + A_LDS_FLOATS;
    float* B1 = B0 + B_LDS_FLOATS;

    // Stage A panel (8 x b128 per thread) and first weight block (4 x b128).
#pragma unroll
    for (int q = 0; q < 8; ++q) {
        const int i4 = (q * 256 + t) * 4;
        lds_copy_f4(Asrc + i4, Al + i4);
    }
#pragma unroll
    for (int q = 0; q < 4; ++q) {
        const int i4 = (q * 256 + t) * 4;
        lds_copy_f4(W + i4, B0 + i4);
    }
    ASYNC_WAIT(0);
    __syncthreads();

    for (int kb = 0; kb < NKB; ++kb) {
        float* Bcur = (kb & 1) ? B1 : B0;
        float* Bnxt = (kb & 1) ? B0 : B1;

        // Prefetch next weight block while computing on the current one.
        if (kb + 1 < NKB) {
            const float* Wsrc = W + (size_t)(kb + 1) * KB * GD;
#pragma unroll
            for (int q = 0; q < 4; ++q) {
                const int i4 = (q * 256 + t) * 4;
                lds_copy_f4(Wsrc + i4, Bnxt + i4);
            }
            ASYNC_WAIT(4);      // allow the 4 just-issued; current block is done
        } else {
            ASYNC_WAIT(0);
        }
        __syncthreads();

#pragma unroll
        for (int kk = 0; kk < KB / 4; ++kk) {
            const int ka = kb * KB + kk * 4 + 2 * hi;   // absolute K for A
            const int kl = kk * 4 + 2 * hi;             // local K for W block
            v2f af;
            af.x = Al[lr * GD + ka];
            af.y = Al[lr * GD + ka + 1];
#pragma unroll
            for (int tt = 0; tt < 4; ++tt) {
                const int n = n0 + tt * 16 + lr;
                v2f bf;
                bf.x = Bcur[kl * GD + n];
                bf.y = Bcur[(kl + 1) * GD + n];
                acc[tt] = __builtin_amdgcn_wmma_f32_16x16x4_f32(
                    false, af, false, bf, (short)0, acc[tt], false, false);
            }
        }
        __syncthreads();   // all waves done with Bcur before it is overwritten
    }
}

// ---------------------------------------------------------------------------
// Kernel 1: A = H @ W1a, Bm = H @ W1b   (grid.y picks which)
// grid = (BN/16, 2), block = 256
// ---------------------------------------------------------------------------
__global__ void __launch_bounds__(256)
gemm_dual(const float* __restrict__ H,
          const float* __restrict__ Wa,
          const float* __restrict__ Wb,
          float* __restrict__ Aout,
          float* __restrict__ Bout) {
    __shared__ float lds[LDS_TOTAL];
    const int m0 = blockIdx.x * 16;
    const float* __restrict__ W = (blockIdx.y == 0) ? Wa : Wb;
    float* __restrict__ O       = (blockIdx.y == 0) ? Aout : Bout;

    v8f acc[4] = {};
    gemm_stripe(H + (size_t)m0 * GD, W, lds, acc);

    const int wave = threadIdx.x >> 5;
    const int lane = threadIdx.x & 31;
    const int lr = lane & 15;
    const int hi = lane >> 4;
#pragma unroll
    for (int tt = 0; tt < 4; ++tt) {
        const int n = wave * 64 + tt * 16 + lr;
#pragma unroll
        for (int v = 0; v < 8; ++v) {
            O[(size_t)(m0 + v + 8 * hi) * GD + n] = acc[tt][v];
        }
    }
}

// ---------------------------------------------------------------------------
// Kernel 2: Zsum[b,i,d] = sum_j adj[b,i,j] * relu(A[b,i,d] + Bm[b,j,d] + b1[d])
//           deg[b,i]    = sum_j adj[b,i,j]
// grid = BN, block = 512 (one d per thread); adj test is wave-uniform.
// ---------------------------------------------------------------------------
__global__ void __launch_bounds__(GD)
reduce_msg(const float* __restrict__ Am,
           const float* __restrict__ Bm,
           const float* __restrict__ adj,
           const float* __restrict__ b1,
           float* __restrict__ Zsum,
           float* __restrict__ deg) {
    const int row = blockIdx.x;
    const int b   = row / GN;
    const int i   = row - b * GN;
    const int d   = threadIdx.x;

    const float areg = Am[(size_t)row * GD + d] + b1[d];
    const float* __restrict__ adjrow = adj + (size_t)(b * GN + i) * GN;
    const float* __restrict__ bbase  = Bm + (size_t)(b * GN) * GD + d;

    float acc = 0.0f;
    float dacc = 0.0f;
#pragma unroll 4
    for (int j = 0; j < GN; ++j) {
        const float av = adjrow[j];          // uniform across the block
        dacc += av;
        if (av != 0.0f) {
            acc += av * fmaxf(areg + bbase[(size_t)j * GD], 0.0f);
        }
    }
    Zsum[(size_t)row * GD + d] = acc;
    if (threadIdx.x == 0) deg[row] = dacc;
}

// ---------------------------------------------------------------------------
// Kernel 3: msg = inv_n*(Zsum @ W2) + inv_n*deg*b2 ; h' = LayerNorm(h + msg)
// grid = BN/16, block = 256. GEMM core + fused LN over the 16x512 LDS tile
// (reuses the A-panel LDS region). Wave32 shuffle reductions; conflict-free
// strided LDS access (e = q*32 + lane).
// ---------------------------------------------------------------------------
__global__ void __launch_bounds__(256)
gemm2_ln(const float* __restrict__ Z,
         const float* __restrict__ W2,
         const float* __restrict__ b2,
         const float* __restrict__ deg,
         const float* __restrict__ Hin,
         const float* __restrict__ gamma,
         const float* __restrict__ beta,
         float* __restrict__ Hout,
         int do_clip) {
    __shared__ float lds[LDS_TOTAL];
    const int m0 = blockIdx.x * 16;

    v8f acc[4] = {};
    gemm_stripe(Z + (size_t)m0 * GD, W2, lds, acc);

    const int wave = threadIdx.x >> 5;
    const int lane = threadIdx.x & 31;
    const int lr = lane & 15;
    const int hi = lane >> 4;
    float* tile = lds;                       // reuse A region: 16 x 512

    // Epilogue: scale, bias, residual -> LDS tile
#pragma unroll
    for (int tt = 0; tt < 4; ++tt) {
        const int e = wave * 64 + tt * 16 + lr;
        const float b2e = b2[e] * G_INVN;
#pragma unroll
        for (int v = 0; v < 8; ++v) {
            const int r   = v + 8 * hi;      // 0..15
            const int row = m0 + r;
            tile[r * GD + e] = acc[tt][v] * G_INVN + deg[row] * b2e
                             + Hin[(size_t)row * GD + e];
        }
    }
    __syncthreads();

    // LayerNorm: each wave owns 2 rows; lanes stride by 32 (bank-conflict free)
#pragma unroll
    for (int rr = 0; rr < 2; ++rr) {
        const int r = wave * 2 + rr;
        float s = 0.0f, s2 = 0.0f;
#pragma unroll
        for (int q = 0; q < 16; ++q) {
            const float x = tile[r * GD + q * 32 + lane];
            s += x;
            s2 += x * x;
        }
#pragma unroll
        for (int off = 16; off > 0; off >>= 1) {
            s  += __shfl_xor(s,  off, 32);
            s2 += __shfl_xor(s2, off, 32);
        }
        const float mu  = s * (1.0f / (float)GD);
        const float var = s2 * (1.0f / (float)GD) - mu * mu;
        const float inv = rsqrtf(var + G_EPS);
        const int row = m0 + r;
#pragma unroll
        for (int q = 0; q < 16; ++q) {
            const int e = q * 32 + lane;
            float y = (tile[r * GD + e] - mu) * inv * gamma[e] + beta[e];
            if (do_clip) y = fminf(fmaxf(y, -100.0f), 100.0f);
            Hout[(size_t)row * GD + e] = y;
        }
    }
}

// ---------------------------------------------------------------------------
// Host launcher
// ---------------------------------------------------------------------------
extern "C" void kernel_launch(void* const* d_in, const int* in_sizes, int n_in,
                              void* d_out, int out_size, void* d_ws, size_t ws_size,
                              hipStream_t stream) {
    (void)in_sizes; (void)n_in; (void)out_size; (void)ws_size;

    const float* H0    = (const float*)d_in[0];   // (B,N,D)
    const float* adj   = (const float*)d_in[1];   // (B,N,N)
    const float* W1a   = (const float*)d_in[2];   // (L,D,D)
    const float* W1b   = (const float*)d_in[3];   // (L,D,D)
    const float* b1    = (const float*)d_in[4];   // (L,D)
    const float* W2    = (const float*)d_in[5];   // (L,D,D)
    const float* b2    = (const float*)d_in[6];   // (L,D)
    const float* gamma = (const float*)d_in[7];   // (D,)
    const float* beta  = (const float*)d_in[8];   // (D,)
    float* out = (float*)d_out;

    float* ws  = (float*)d_ws;
    float* A   = ws;                          // BN*D
    float* Bm  = A  + (size_t)GBN * GD;       // BN*D
    float* Zs  = Bm + (size_t)GBN * GD;       // BN*D
    float* Hb  = Zs + (size_t)GBN * GD;       // BN*D (h after layer 0)
    float* deg = Hb + (size_t)GBN * GD;       // BN

    for (int l = 0; l < GL; ++l) {
        const float* Hin = (l == 0) ? H0 : Hb;
        float* Hout      = (l == GL - 1) ? out : Hb;
        const size_t wo  = (size_t)l * GD * GD;

        gemm_dual<<<dim3(GBN / 16, 2), 256, 0, stream>>>(
            Hin, W1a + wo, W1b + wo, A, Bm);

        reduce_msg<<<GBN, GD, 0, stream>>>(
            A, Bm, adj, b1 + (size_t)l * GD, Zs, deg);

        gemm2_ln<<<GBN / 16, 256, 0, stream>>>(
            Zs, W2 + wo, b2 + (size_t)l * GD, deg, Hin, gamma, beta,
            Hout, (l == GL - 1) ? 1 : 0);
    }
}